// DRModule_34815004901652
// MI455X (gfx1250) — compile-verified
//
#include <hip/hip_runtime.h>

typedef __attribute__((ext_vector_type(2))) float v2f;
typedef __attribute__((ext_vector_type(8))) float v8f;

#define B_   32
#define C_   256
#define H_   64
#define W_   64
#define CO_  512        // 2*C
#define KTOT 2304       // C_*9
#define HW_  4096

// pass-1 tiling
#define M_TILE 64       // output channels per block
#define N_TILE 128      // pixels per block = 2 rows x 64
#define KB     8        // input channels staged per iteration
#define NSTAGE (C_ / KB)

// LDS layout (floats)
// A: per co row, [s][cib] transposed slab: offset = co*A_PITCH + s*8 + cib
#define A_PITCH      74                     // 72 data + 2 pad
#define A_FLOATS     (M_TILE * A_PITCH)     // 4736
// B: channel-pair interleaved: offset = cp*B_CP_PITCH + row*B_ROW_PITCH + pos*2 + e
//    pos: [1]=x=-1 halo, [2..65]=x0..63, [66]=x=64 halo   (cp = cib>>1, e = cib&1)
#define B_ROW_PITCH  136                    // 68 positions * 2 interleaved channels
#define B_CP_PITCH   (4 * B_ROW_PITCH)      // 544
#define B_FLOATS     (4 * B_CP_PITCH)       // 2176

// generic pointer to __shared__ -> LDS byte offset (low 32 bits, LDS aperture rule)
__device__ __forceinline__ unsigned lds_off(const void* p) {
    return (unsigned)(uintptr_t)p;
}

// CDNA5 async scatter copy: global -> LDS, 4 bytes per lane, tracked by ASYNCcnt
__device__ __forceinline__ void async_copy_b32(unsigned lds, unsigned long long gaddr) {
    asm volatile("global_load_async_to_lds_b32 %0, %1, off"
                 :: "v"(lds), "v"(gaddr) : "memory");
}

// Issue one K-stage of async copies: 18 (A, transposed) + 8 (B, interleaved) = 26 ops.
__device__ __forceinline__ void issue_stage(
    const float* __restrict__ x, const float* __restrict__ w,
    int b, int y0, int co0, int ci0, int tid,
    unsigned aBase, unsigned bBase, unsigned trashBase)
{
    // A: w_off[co0+co_l][ci0..ci0+7][0..8] -> Alds[co_l][s][cib]
    {
        const int co_l = tid >> 2;
        const int q    = tid & 3;          // 18-element chunk: cib pair {2q, 2q+1}
        const float* gsrc = w + (size_t)(co0 + co_l) * KTOT + ci0 * 9 + q * 18;
        const unsigned lbase = aBase + (unsigned)(co_l * A_PITCH * 4);
        #pragma unroll
        for (int e = 0; e < 18; ++e) {
            const int cib = q * 2 + e / 9;
            const int s   = e % 9;
            async_copy_b32(lbase + (unsigned)((s * 8 + cib) * 4),
                           (unsigned long long)(uintptr_t)(gsrc + e));
        }
    }
    // B: x[b][ci0+cl][y0-1..y0+2][0..63] -> interleaved pairs; OOB rows -> trash
    {
        const int rid  = tid >> 3;
        const int cl   = rid >> 2;
        const int rr   = rid & 3;
        const int col0 = (tid & 7) * 8;
        const int y    = y0 - 1 + rr;
        const int yc   = min(max(y, 0), H_ - 1);
        const bool ok  = (y == yc);
        const float* gsrc = x + (size_t)(b * C_ + ci0 + cl) * HW_ + yc * W_ + col0;
        const unsigned lReal = bBase + (unsigned)
            (((cl >> 1) * B_CP_PITCH + rr * B_ROW_PITCH + (2 + col0) * 2 + (cl & 1)) * 4);
        const unsigned lTrash = trashBase + (unsigned)((tid & 31) * 8);
        #pragma unroll
        for (int i = 0; i < 8; ++i) {
            const unsigned l = ok ? (lReal + 8u * i) : (lTrash + 4u * (i & 1));
            async_copy_b32(l, (unsigned long long)(uintptr_t)(gsrc + i));
        }
    }
}

__global__ __launch_bounds__(256) void conv_offsets_wmma(
    const float* __restrict__ x, const float* __restrict__ w,
    float* __restrict__ off)
{
    __shared__ float Alds[2][A_FLOATS];   // 2 x 18.5 KB
    __shared__ float Blds[2][B_FLOATS];   // 2 x  8.5 KB
    __shared__ float Trash[128];          // dump for OOB async lanes

    const int tid     = threadIdx.x;
    const int lane    = tid & 31;
    const int wid     = tid >> 5;
    const int wave_m  = wid & 1;     // 2 waves along M (32 co each)
    const int wave_n  = wid >> 1;    // 4 waves along N (32 px each)
    const int lane_lo = lane & 15;
    const int lane_hi = lane >> 4;

    const int n0  = blockIdx.x * N_TILE;
    const int b   = n0 >> 12;
    const int y0  = (n0 & (HW_ - 1)) >> 6;
    const int co0 = blockIdx.y * M_TILE;

    const unsigned aBase0 = lds_off(&Alds[0][0]);
    const unsigned aBase1 = lds_off(&Alds[1][0]);
    const unsigned bBase0 = lds_off(&Blds[0][0]);
    const unsigned bBase1 = lds_off(&Blds[1][0]);
    const unsigned tBase  = lds_off(&Trash[0]);

    // zero B buffers once: halo cols + OOB halo rows stay zero forever
    for (int k = tid; k < 2 * B_FLOATS; k += 256)
        (&Blds[0][0])[k] = 0.f;
    __syncthreads();

    v8f acc[2][2];
    #pragma unroll
    for (int mi = 0; mi < 2; ++mi)
        #pragma unroll
        for (int ni = 0; ni < 2; ++ni)
            acc[mi][ni] = (v8f){0.f,0.f,0.f,0.f,0.f,0.f,0.f,0.f};

    static const int DY[9] = {-1,-1,-1, 0,0,0, 1,1,1};
    static const int DX[9] = {-1, 0, 1,-1,0,1,-1,0,1};

    // prologue: stage 0 into buffer 0
    issue_stage(x, w, b, y0, co0, 0, tid, aBase0, bBase0, tBase);

    for (int st = 0; st < NSTAGE; ++st) {
        const int buf = st & 1;
        if (st + 1 < NSTAGE) {
            issue_stage(x, w, b, y0, co0, (st + 1) * KB, tid,
                        buf ? aBase0 : aBase1, buf ? bBase0 : bBase1, tBase);
            asm volatile("s_wait_asynccnt 0x1a" ::: "memory");  // 26 = next stage in flight
        } else {
            asm volatile("s_wait_asynccnt 0x0" ::: "memory");
        }
        __syncthreads();

        const float* Ab = Alds[buf];
        const float* Bb = Blds[buf];

        for (int s = 0; s < 9; ++s) {
            const int dy = DY[s], dx = DX[s];
            #pragma unroll
            for (int kk = 0; kk < KB; kk += 4) {
                const int kb = kk + lane_hi * 2;   // even: contiguous b64 pair (kb, kb+1)
                // A frag: lane M=lane_lo, K = kb + v -> one aligned v2f load
                v2f afr[2];
                #pragma unroll
                for (int mi = 0; mi < 2; ++mi) {
                    const int co_l = wave_m * 32 + mi * 16 + lane_lo;
                    afr[mi] = *(const v2f*)(Ab + co_l * A_PITCH + s * 8 + kb);
                }
                // B frag: lane N=lane_lo, K = kb + v -> one aligned v2f load (interleaved pair)
                v2f bfr[2];
                #pragma unroll
                for (int ni = 0; ni < 2; ++ni) {
                    const int p    = wave_n * 32 + ni * 16 + lane_lo;
                    const int r    = p >> 6;
                    const int xcol = p & 63;
                    bfr[ni] = *(const v2f*)(Bb + (kb >> 1) * B_CP_PITCH
                                               + (r + 1 + dy) * B_ROW_PITCH
                                               + (xcol + dx + 2) * 2);
                }
                #pragma unroll
                for (int mi = 0; mi < 2; ++mi)
                    #pragma unroll
                    for (int ni = 0; ni < 2; ++ni)
                        acc[mi][ni] = __builtin_amdgcn_wmma_f32_16x16x4_f32(
                            false, afr[mi], false, bfr[ni],
                            (short)0, acc[mi][ni], false, false);
            }
        }
        __syncthreads();   // protect buf from stage st+2 overwrite
    }

    // store D: row v + 8*lane_hi, col lane_lo (ISA 16x16 f32 C/D layout)
    #pragma unroll
    for (int mi = 0; mi < 2; ++mi) {
        #pragma unroll
        for (int ni = 0; ni < 2; ++ni) {
            const int p    = wave_n * 32 + ni * 16 + lane_lo;
            const int yy   = y0 + (p >> 6);
            const int xcol = p & 63;
            #pragma unroll
            for (int v = 0; v < 8; ++v) {
                const int co = co0 + wave_m * 32 + mi * 16 + v + 8 * lane_hi;
                off[(size_t)(b * CO_ + co) * HW_ + yy * W_ + xcol] = acc[mi][ni][v];
            }
        }
    }
}

// Pass 2: scrambled-view offsets -> clamp -> bilinear gather -> +residual.
// Pair for (n,i,j) lives at off[n*8192 + i*128 + j*2] of the NCHW conv output.
__global__ __launch_bounds__(256) void deform_sample(
    const float* __restrict__ x, const float* __restrict__ off,
    float* __restrict__ out)
{
    const int idx = blockIdx.x * 256 + threadIdx.x;   // < 2^25
    const int n   = idx >> 12;
    const int rem = idx & (HW_ - 1);
    const int i   = rem >> 6;
    const int j   = rem & 63;

    const float* ofs = off + (size_t)n * (2 * HW_) + i * 128 + j * 2;
    float ci = ofs[0] + (float)i;
    float cj = ofs[1] + (float)j;
    ci = fminf(fmaxf(ci, 0.f), 63.f);
    cj = fminf(fmaxf(cj, 0.f), 63.f);

    const float i0f = floorf(ci), i1f = ceilf(ci);
    const float j0f = floorf(cj), j1f = ceilf(cj);
    const float t0 = ci - i0f;
    const float t1 = cj - j0f;

    const int i0 = (int)i0f, i1 = (int)i1f, j0 = (int)j0f, j1 = (int)j1f;
    const float* xf = x + (size_t)n * HW_;
    const float v_lt = xf[i0 * W_ + j0];
    const float v_rt = xf[i1 * W_ + j0];
    const float v_lb = xf[i0 * W_ + j1];
    const float v_rb = xf[i1 * W_ + j1];
    const float vt = v_lt + (v_rt - v_lt) * t0;
    const float vb = v_lb + (v_rb - v_lb) * t0;
    const float val = vt + (vb - vt) * t1;

    out[idx] = val + x[idx];
}

extern "C" void kernel_launch(void* const* d_in, const int* in_sizes, int n_in,
                              void* d_out, int out_size, void* d_ws, size_t ws_size,
                              hipStream_t stream) {
    const float* x = (const float*)d_in[0];   // (32,256,64,64) f32
    const float* w = (const float*)d_in[1];   // (512,256,3,3)  f32
    float* out = (float*)d_out;               // (32,256,64,64) f32
    float* off = (float*)d_ws;                // (32,512,64,64) f32 scratch = 256 MB

    dim3 g1(B_ * H_ * W_ / N_TILE, CO_ / M_TILE);   // (1024, 8)
    conv_offsets_wmma<<<g1, dim3(256), 0, stream>>>(x, w, off);

    const int total = B_ * C_ * H_ * W_;            // 33554432
    deform_sample<<<dim3(total / 256), dim3(256), 0, stream>>>(x, off, out);
}